// LlamaAttentionTopk_62758062129280
// MI455X (gfx1250) — compile-verified
//
#include <hip/hip_runtime.h>
#include <math.h>

typedef __attribute__((ext_vector_type(2))) float v2f;
typedef __attribute__((ext_vector_type(8))) float v8f;
typedef __attribute__((ext_vector_type(4))) float v4f;

#define BB 4
#define HQ 32
#define HKV 8
#define GROUPS 4
#define DD 128
#define S_PAST 8192
#define HID 4096
#define TOKEN_BUDGET 256
#define INIT_BUDGET 4
#define RECENT_START 7937      // kv_len - RECENT_BUDGET = 8193-256
#define TOPK_LO 4              // INIT_BUDGET
#define TOPK_HI 7936           // kv_len - q_len - RECENT_BUDGET
#define SCALE 0.08838834764831845f  // 1/sqrt(128)

// monotonic float->uint key: a >= b  <=>  fkey(a) >= fkey(b)
__device__ __forceinline__ unsigned fkey(float f) {
  unsigned u = __float_as_uint(f);
  return (u & 0x80000000u) ? ~u : (u | 0x80000000u);
}

// ---------------------------------------------------------------------------
// Kernel 1: fused QKV GEMV + RoPE.
// grid = B*24 blocks, 256 threads. Concat column space: [0,4096)=Q,
// [4096,5120)=Knew, [5120,6144)=Vnew. Hidden row cached in LDS; weight reads
// fully coalesced (thread t -> column t).
// ---------------------------------------------------------------------------
__global__ __launch_bounds__(256) void qkv_rope_kernel(
    const float* __restrict__ hidden, const float* __restrict__ cosp,
    const float* __restrict__ sinp, const float* __restrict__ wq,
    const float* __restrict__ wk, const float* __restrict__ wv,
    float* __restrict__ q_ws, float* __restrict__ knew,
    float* __restrict__ vnew) {
  __shared__ float hs[HID];
  __shared__ float raw[256];
  const int b = blockIdx.x / 24;
  const int seg = blockIdx.x % 24;
  const int tid = threadIdx.x;

  for (int k = tid; k < HID; k += 256) hs[k] = hidden[(size_t)b * HID + k];
  __syncthreads();

  const int col = seg * 256 + tid;
  const float* W;
  int N, c;
  if (col < 4096)      { W = wq; N = 4096; c = col; }
  else if (col < 5120) { W = wk; N = 1024; c = col - 4096; }
  else                 { W = wv; N = 1024; c = col - 5120; }

  float acc = 0.f;
#pragma unroll 4
  for (int k = 0; k < HID; ++k) acc += hs[k] * W[(size_t)k * N + c];

  raw[tid] = acc;
  __syncthreads();

  float outv;
  if (col >= 5120) {  // V: no RoPE
    outv = acc;
  } else {            // Q or K: RoPE (block = exactly 2 heads of 128)
    const int d = tid & 127;
    const float partner = (d < 64) ? -raw[tid + 64] : raw[tid - 64];
    const float cs = cosp[(size_t)b * DD + d];
    const float sn = sinp[(size_t)b * DD + d];
    outv = acc * cs + partner * sn;
  }

  if (col < 4096)      q_ws[(size_t)b * 4096 + col] = outv;
  else if (col < 5120) knew[(size_t)b * 1024 + (col - 4096)] = outv;
  else                 vnew[(size_t)b * 1024 + (col - 5120)] = outv;
}

// ---------------------------------------------------------------------------
// Kernel 2: attention scores via V_WMMA_F32_16X16X4_F32.
// grid = (16 chunks, B*HKV), 256 threads = 8 waves. Each wave owns a 16-key
// tile; M = keys, N = the 4 q-heads of the GQA group (cols 0..3 used),
// K = 128 accumulated in 32 chained wmma steps of 4.
// A-lane layout (16x4 f32): lane L<16 -> row M=L holds {K=0,K=1}; lane L+16
// holds {K=2,K=3}. B mirrors this with rows = d, cols = q-head.
// ---------------------------------------------------------------------------
__global__ __launch_bounds__(256) void scores_kernel(
    const float* __restrict__ past_key, const float* __restrict__ q_ws,
    float* __restrict__ scores) {
  __shared__ float qs[16 * DD];  // padded q-group (rows 0..3 real, scaled)
  const int b = blockIdx.y >> 3;
  const int kv = blockIdx.y & 7;
  const int tid = threadIdx.x;
  const int wave = tid >> 5;
  const int lane = tid & 31;
  const int m = lane & 15;
  const int hi = lane >> 4;

  // fill q tile: rows 0..3 = scaled q of heads kv*4+row, rows 4..15 = 0
  for (int i = tid; i < 16 * DD; i += 256) {
    const int row = i >> 7, d = i & 127;
    qs[i] = (row < 4) ? q_ws[((size_t)b * HQ + kv * GROUPS + row) * DD + d] * SCALE
                      : 0.f;
  }
  __syncthreads();

  const float* kbase = past_key + ((size_t)b * HKV + kv) * S_PAST * DD;

  for (int it = 0; it < 4; ++it) {
    const int key_base = blockIdx.x * 512 + it * 128 + wave * 16;
    const float* krow = kbase + (size_t)key_base * DD;

    // prefetch this lane's key row for the next block-iteration
    if (key_base + 144 <= S_PAST)
      __builtin_prefetch(krow + (size_t)128 * DD + (size_t)m * DD, 0, 1);

    v8f c = {};
#pragma unroll
    for (int s = 0; s < 32; ++s) {
      const int d = 4 * s + 2 * hi;
      v2f a = *(const v2f*)(krow + (size_t)m * DD + d);   // A: key tile
      v2f q2 = *(const v2f*)(&qs[m * DD + d]);            // B: q-group^T
      c = __builtin_amdgcn_wmma_f32_16x16x4_f32(
          false, a, false, q2, (short)0, c, false, false);
    }

    // D layout: VGPR r -> (M=r, N=lane) lanes 0-15, (M=r+8, N=lane-16) 16-31.
    if ((lane & 15) < 4) {
      const int n = lane & 15;  // q-head within group
      float* srow = scores + ((size_t)b * HQ + kv * GROUPS + n) * S_PAST +
                    key_base + hi * 8;
#pragma unroll
      for (int r = 0; r < 8; ++r) srow[r] = c[r];
    }
  }
}

// ---------------------------------------------------------------------------
// Kernel 3: per-(b,h) radix-select: find key of the 256th-largest score in
// positions [4,7936]. 4 passes of 8-bit histograms (MSB first) in LDS.
// grid = B*HQ, 256 threads.
// ---------------------------------------------------------------------------
__global__ __launch_bounds__(256) void topk_kernel(
    const float* __restrict__ scores, unsigned* __restrict__ thr) {
  __shared__ unsigned hist[256];
  __shared__ unsigned sh_prefix, sh_rem;
  const int bh = blockIdx.x;
  const int tid = threadIdx.x;
  const float* srow = scores + (size_t)bh * S_PAST;

  if (tid == 0) { sh_prefix = 0u; sh_rem = TOKEN_BUDGET; }
  __syncthreads();

  for (int shift = 24; shift >= 0; shift -= 8) {
    hist[tid] = 0u;
    __syncthreads();
    const unsigned prefix = sh_prefix;
    for (int pos = TOPK_LO + tid; pos <= TOPK_HI; pos += 256) {
      const unsigned k = fkey(srow[pos]);
      if (shift == 24 || ((k ^ prefix) >> (shift + 8)) == 0u)
        atomicAdd(&hist[(k >> shift) & 255u], 1u);
    }
    __syncthreads();
    if (tid == 0) {
      unsigned rem = sh_rem;
      int bin = 255;
      for (; bin > 0; --bin) {
        if (hist[bin] >= rem) break;
        rem -= hist[bin];
      }
      sh_prefix = prefix | ((unsigned)bin << shift);
      sh_rem = (rem == 0u) ? 1u : rem;
    }
    __syncthreads();
  }
  if (tid == 0) thr[bh] = sh_prefix;  // keep iff fkey(score) >= thr
}

// ---------------------------------------------------------------------------
// Kernel 4: sparse softmax + A·V gather. grid = B*HQ, 128 threads (thread t
// owns dim d=t). Builds kept-position list in LDS (~516 entries), then
// streams only kept value rows (512B coalesced per row).
// ---------------------------------------------------------------------------
__global__ __launch_bounds__(128) void softmax_av_kernel(
    const float* __restrict__ scores, const unsigned* __restrict__ thr,
    const float* __restrict__ q_ws, const float* __restrict__ knew,
    const float* __restrict__ vnew, const float* __restrict__ past_value,
    float* __restrict__ attn) {
  __shared__ int list[640];
  __shared__ int cnt;
  __shared__ float red[128];
  const int bh = blockIdx.x;
  const int b = bh >> 5;
  const int h = bh & 31;
  const int kv = h >> 2;
  const int tid = threadIdx.x;
  const float* srow = scores + (size_t)bh * S_PAST;
  const unsigned tkey = thr[bh];

  if (tid == 0) cnt = 0;
  __syncthreads();

  for (int pos = tid; pos < S_PAST; pos += 128) {
    bool keep = (pos < INIT_BUDGET) || (pos >= RECENT_START);
    if (!keep) keep = fkey(srow[pos]) >= tkey;  // pos in [4,7936]
    if (keep) {
      const int i = atomicAdd(&cnt, 1);
      if (i < 640) list[i] = pos;
    }
  }
  __syncthreads();
  const int n = (cnt < 640) ? cnt : 640;

  // score of the new key (pos 8192, always kept): dot(q, k_new) * scale
  red[tid] = q_ws[((size_t)b * HQ + h) * DD + tid] *
             knew[((size_t)b * HKV + kv) * DD + tid];
  __syncthreads();
  for (int off = 64; off > 0; off >>= 1) {
    if (tid < off) red[tid] += red[tid + off];
    __syncthreads();
  }
  const float s_new = red[0] * SCALE;
  __syncthreads();

  // pass 1: max (all threads compute the identical reduction; broadcast loads)
  float mx = s_new;
  for (int i = 0; i < n; ++i) mx = fmaxf(mx, srow[list[i]]);

  // pass 2: weights + gathered value accumulation along dim d = tid
  const float* vbase = past_value + ((size_t)b * HKV + kv) * S_PAST * DD;
  float acc = 0.f, sum = 0.f;
  for (int i = 0; i < n; ++i) {
    const int pos = list[i];
    const float w = __expf(srow[pos] - mx);
    sum += w;
    acc += w * vbase[(size_t)pos * DD + tid];
  }
  const float wn = __expf(s_new - mx);
  sum += wn;
  acc += wn * vnew[((size_t)b * HKV + kv) * DD + tid];

  attn[(size_t)b * 4096 + h * DD + tid] = acc / sum;
}

// ---------------------------------------------------------------------------
// Kernel 5: output projection GEMV: out = attn @ wo. grid = B*16, 256 thr.
// ---------------------------------------------------------------------------
__global__ __launch_bounds__(256) void out_proj_kernel(
    const float* __restrict__ attn, const float* __restrict__ wo,
    float* __restrict__ out) {
  __shared__ float as[4096];
  const int b = blockIdx.x >> 4;
  const int seg = blockIdx.x & 15;
  const int tid = threadIdx.x;

  for (int k = tid; k < 4096; k += 256) as[k] = attn[(size_t)b * 4096 + k];
  __syncthreads();

  const int col = seg * 256 + tid;
  float acc = 0.f;
#pragma unroll 4
  for (int k = 0; k < 4096; ++k) acc += as[k] * wo[(size_t)k * 4096 + col];
  out[(size_t)b * 4096 + col] = acc;
}

// ---------------------------------------------------------------------------
extern "C" void kernel_launch(void* const* d_in, const int* in_sizes, int n_in,
                              void* d_out, int out_size, void* d_ws,
                              size_t ws_size, hipStream_t stream) {
  const float* hidden     = (const float*)d_in[0];
  const float* cosp       = (const float*)d_in[1];
  const float* sinp       = (const float*)d_in[2];
  const float* past_key   = (const float*)d_in[3];
  const float* past_value = (const float*)d_in[4];
  const float* wq         = (const float*)d_in[5];
  const float* wk         = (const float*)d_in[6];
  const float* wv         = (const float*)d_in[7];
  const float* wo         = (const float*)d_in[8];
  float* out = (float*)d_out;

  float* ws = (float*)d_ws;
  float* q_ws   = ws;               // 16384 floats: [B][HQ][D]
  float* knew   = ws + 16384;       //  4096 floats: [B][HKV][D] (roped)
  float* vnew   = ws + 20480;       //  4096 floats
  float* attn   = ws + 24576;       // 16384 floats: [B][HQ*D]
  unsigned* thr = (unsigned*)(ws + 40960);  // 128
  float* scores = ws + 41088;       // B*HQ*S_PAST = 1,048,576 floats

  qkv_rope_kernel<<<dim3(BB * 24), 256, 0, stream>>>(
      hidden, cosp, sinp, wq, wk, wv, q_ws, knew, vnew);
  scores_kernel<<<dim3(16, BB * HKV), 256, 0, stream>>>(past_key, q_ws, scores);
  topk_kernel<<<dim3(BB * HQ), 256, 0, stream>>>(scores, thr);
  softmax_av_kernel<<<dim3(BB * HQ), 128, 0, stream>>>(
      scores, thr, q_ws, knew, vnew, past_value, attn);
  out_proj_kernel<<<dim3(BB * 16), 256, 0, stream>>>(attn, wo, out);
}